// PermutedNetwork_2_63333587747098
// MI455X (gfx1250) — compile-verified
//
#include <hip/hip_runtime.h>
#include <hip/hip_bf16.h>

typedef float v2f __attribute__((ext_vector_type(2)));
typedef float v8f __attribute__((ext_vector_type(8)));

// ---------------------------------------------------------------------------
// Plain 3x3 conv (stride 1, pad 1), OC fixed at 18 (offset branch).
// ---------------------------------------------------------------------------
template <int C, int H, int W>
__global__ void conv3x3_off_kernel(const float* __restrict__ in,
                                   const float* __restrict__ w,
                                   const float* __restrict__ bias,
                                   float* __restrict__ out, int Bt) {
  int idx = blockIdx.x * blockDim.x + threadIdx.x;
  int total = Bt * 18 * H * W;
  if (idx >= total) return;
  int x = idx % W;
  int y = (idx / W) % H;
  int oc = (idx / (W * H)) % 18;
  int b = idx / (W * H * 18);
  float acc = bias[oc];
  for (int c = 0; c < C; ++c) {
    const float* ip = in + (size_t)(b * C + c) * H * W;
    const float* wp = w + (oc * C + c) * 9;
#pragma unroll
    for (int ky = 0; ky < 3; ++ky) {
      int yy = y + ky - 1;
      if (yy < 0 || yy >= H) continue;
#pragma unroll
      for (int kx = 0; kx < 3; ++kx) {
        int xx = x + kx - 1;
        if (xx < 0 || xx >= W) continue;
        acc = fmaf(ip[yy * W + xx], wp[ky * 3 + kx], acc);
      }
    }
  }
  out[idx] = acc;
}

// ---------------------------------------------------------------------------
// Deformable conv layer as WMMA GEMM.
// One wave (32 threads) handles a tile of 16 output positions (N=16) for all
// O output channels (O/16 M-tiles). Bilinear corner setup + patch staged in
// LDS, contraction via V_WMMA_F32_16X16X4_F32 (full fp32 precision).
// A fragments come from branch-free aligned v2f loads: global b64 when K9 is
// already a multiple of 4 (layers 2/3), or an LDS-staged zero-padded weight
// copy (layer 1, K9=9 -> KP=12). B fragments are contiguous ds b64 reads.
// Applies bias + ReLU on store.
// ---------------------------------------------------------------------------
template <int C, int O, int H, int W>
__global__ void __launch_bounds__(32)
deform_wmma_kernel(const float* __restrict__ xin,
                   const float* __restrict__ off,
                   const float* __restrict__ w,
                   const float* __restrict__ bias,
                   float* __restrict__ out) {
  constexpr int K9 = C * 9;                // true K
  constexpr int KP = (K9 + 3) & ~3;       // K padded to multiple of 4
  constexpr bool PADK = (KP != K9);       // needs zero-padded weight staging
  constexpr int NT = O / 16;              // number of 16-wide M tiles
  constexpr int HW = H * W;

  __shared__ float patch[16][KP];         // B tile, transposed: [n][k]
  __shared__ float cw[9][16][4];          // bilinear corner weights (0 if OOB)
  __shared__ int   ci[9][16][4];          // clamped corner flat indices
  __shared__ float wpad[PADK ? (O * KP) : 1];  // padded weights (layer 1)

  const int lane = threadIdx.x;
  const int pos0 = blockIdx.x * 16;       // first of 16 positions (same image)
  const int b = pos0 / HW;
  const int sp0 = pos0 % HW;

  // ---- Stage 0: zero-padded weight staging (only when K9 % 4 != 0) ------
  if (PADK) {
    for (int t = lane; t < O * KP; t += 32) {
      int m = t / KP;
      int kk = t % KP;
      wpad[t] = (kk < K9) ? w[m * K9 + kk] : 0.f;
    }
  }

  // ---- Stage 1: bilinear setup per (tap k, position n) -------------------
  for (int e = lane; e < 9 * 16; e += 32) {
    int k = e >> 4;
    int n = e & 15;
    int sp = sp0 + n;
    int yy = sp / W;
    int xx = sp % W;
    float dy = off[((size_t)b * 18 + 2 * k) * HW + sp];
    float dx = off[((size_t)b * 18 + 2 * k + 1) * HW + sp];
    float py = (float)yy + (float)(k / 3 - 1) + dy;
    float px = (float)xx + (float)(k % 3 - 1) + dx;
    float y0f = floorf(py), x0f = floorf(px);
    float fy = py - y0f, fx = px - x0f;
    int y0 = (int)y0f, x0 = (int)x0f;
#pragma unroll
    for (int c4 = 0; c4 < 4; ++c4) {
      int cy = y0 + (c4 >> 1);
      int cx = x0 + (c4 & 1);
      float wt = ((c4 >> 1) ? fy : 1.f - fy) * ((c4 & 1) ? fx : 1.f - fx);
      bool valid = (cy >= 0) && (cy < H) && (cx >= 0) && (cx < W);
      int cyc = cy < 0 ? 0 : (cy > H - 1 ? H - 1 : cy);
      int cxc = cx < 0 ? 0 : (cx > W - 1 ? W - 1 : cx);
      cw[k][n][c4] = valid ? wt : 0.f;
      ci[k][n][c4] = cyc * W + cxc;
    }
  }
  __syncthreads();

  // ---- Stage 2: gather C*9 x 16 patch into LDS (transposed layout) -------
  for (int t = lane; t < K9 * 16; t += 32) {
    int n = t & 15;
    int r = t >> 4;                        // r = c*9 + k
    int c = r / 9;
    int k = r % 9;
    const float* ip = xin + (size_t)(b * C + c) * HW;
    float v = cw[k][n][0] * ip[ci[k][n][0]] + cw[k][n][1] * ip[ci[k][n][1]] +
              cw[k][n][2] * ip[ci[k][n][2]] + cw[k][n][3] * ip[ci[k][n][3]];
    patch[n][r] = v;
  }
  for (int t = K9 * 16 + lane; t < KP * 16; t += 32)
    patch[t & 15][t >> 4] = 0.f;           // zero-pad K columns
  __syncthreads();

  // ---- Stage 3: WMMA contraction  D[MxN] += A[Mx4] * B[4xN] --------------
  v8f zero = {0.f, 0.f, 0.f, 0.f, 0.f, 0.f, 0.f, 0.f};
  v8f acc[NT];
#pragma unroll
  for (int t = 0; t < NT; ++t) acc[t] = zero;

  const int half = lane >> 4;              // 0: K sub 0,1  1: K sub 2,3
  const int row = lane & 15;               // M row (for A), N col (for B)
  for (int k0 = 0; k0 < KP; k0 += 4) {
    int ka = k0 + 2 * half;                // even -> 8B-aligned v2f loads
    v2f bf = *reinterpret_cast<const v2f*>(&patch[row][ka]);
#pragma unroll
    for (int t = 0; t < NT; ++t) {
      int m = t * 16 + row;
      v2f af;
      if (PADK) {
        af = *reinterpret_cast<const v2f*>(&wpad[m * KP + ka]);
      } else {
        af = *reinterpret_cast<const v2f*>(w + m * K9 + ka);
      }
      acc[t] = __builtin_amdgcn_wmma_f32_16x16x4_f32(
          false, af, false, bf, (short)0, acc[t], false, false);
    }
  }

  // ---- Stage 4: bias + ReLU + store (C/D striping: VGPR r -> M=r+8*half) -
  int sp = sp0 + row;
#pragma unroll
  for (int t = 0; t < NT; ++t) {
#pragma unroll
    for (int r8 = 0; r8 < 8; ++r8) {
      int o = t * 16 + half * 8 + r8;
      float v = acc[t][r8] + bias[o];
      out[((size_t)b * O + o) * HW + sp] = fmaxf(v, 0.f);
    }
  }
}

// ---------------------------------------------------------------------------
// 2x2 max pool, stride 2
// ---------------------------------------------------------------------------
__global__ void maxpool_kernel(const float* __restrict__ in,
                               float* __restrict__ out, int BC, int Hin,
                               int Win) {
  int Hout = Hin >> 1, Wout = Win >> 1;
  int idx = blockIdx.x * blockDim.x + threadIdx.x;
  int total = BC * Hout * Wout;
  if (idx >= total) return;
  int x = idx % Wout;
  int y = (idx / Wout) % Hout;
  int bc = idx / (Wout * Hout);
  const float* ip = in + ((size_t)bc * Hin + 2 * y) * Win + 2 * x;
  out[idx] = fmaxf(fmaxf(ip[0], ip[1]), fmaxf(ip[Win], ip[Win + 1]));
}

// ---------------------------------------------------------------------------
// adaptive avg pool [64,64,24,24] -> [64,576]  (3x3 bins of 8x8)
// ---------------------------------------------------------------------------
__global__ void avgpool_kernel(const float* __restrict__ in,
                               float* __restrict__ out) {
  int idx = blockIdx.x * blockDim.x + threadIdx.x;
  if (idx >= 64 * 64 * 9) return;
  int tx = idx % 3;
  int ty = (idx / 3) % 3;
  int c = (idx / 9) % 64;
  int b = idx / (9 * 64);
  const float* ip = in + (size_t)(b * 64 + c) * 576;
  float s = 0.f;
#pragma unroll
  for (int yy = 0; yy < 8; ++yy)
#pragma unroll
    for (int xx = 0; xx < 8; ++xx)
      s += ip[(ty * 8 + yy) * 24 + tx * 8 + xx];
  out[b * 576 + c * 9 + ty * 3 + tx] = s * (1.f / 64.f);
}

// ---------------------------------------------------------------------------
// FC: [64,576] @ [10,576]^T + b -> [64,10]
// ---------------------------------------------------------------------------
__global__ void fc_kernel(const float* __restrict__ pooled,
                          const float* __restrict__ wfc,
                          const float* __restrict__ bfc,
                          float* __restrict__ out) {
  int idx = blockIdx.x * blockDim.x + threadIdx.x;
  if (idx >= 64 * 10) return;
  int j = idx % 10;
  int b = idx / 10;
  float s = bfc[j];
  for (int i = 0; i < 576; ++i)
    s = fmaf(pooled[b * 576 + i], wfc[j * 576 + i], s);
  out[b * 10 + j] = s;
}

extern "C" void kernel_launch(void* const* d_in, const int* in_sizes, int n_in,
                              void* d_out, int out_size, void* d_ws,
                              size_t ws_size, hipStream_t stream) {
  const float* x      = (const float*)d_in[0];
  const float* w_off1 = (const float*)d_in[1];
  const float* b_off1 = (const float*)d_in[2];
  const float* w1     = (const float*)d_in[3];
  const float* b1     = (const float*)d_in[4];
  const float* w_off2 = (const float*)d_in[5];
  const float* b_off2 = (const float*)d_in[6];
  const float* w2     = (const float*)d_in[7];
  const float* b2     = (const float*)d_in[8];
  const float* w_off3 = (const float*)d_in[9];
  const float* b_off3 = (const float*)d_in[10];
  const float* w3     = (const float*)d_in[11];
  const float* b3     = (const float*)d_in[12];
  const float* w_fc   = (const float*)d_in[13];
  const float* b_fc   = (const float*)d_in[14];
  float* out = (float*)d_out;
  float* WS = (float*)d_ws;

  constexpr int B = 64;
  // Ping-pong workspace layout (element offsets); peak usage ~80 MB.
  float* off1   = WS;              // 64*18*96*96 = 10,616,832
  float* h1     = WS + 10616832;   // 64*16*96*96 =  9,437,184
  float* p1     = WS;              // 64*16*48*48 =  2,359,296 (off1 dead)
  float* off2   = WS + 2359296;    // 64*18*48*48 =  2,654,208
  float* h2     = WS + 10616832;   // 64*32*48*48 =  4,718,592 (h1 dead)
  float* p2     = WS;              // 64*32*24*24 =  1,179,648 (p1/off2 dead)
  float* off3   = WS + 1179648;    // 64*18*24*24 =    663,552
  float* h3     = WS + 10616832;   // 64*64*24*24 =  2,359,296 (h2 dead)
  float* pooled = WS;              // 64*576 (p2/off3 dead)

  // Stage 1: 96x96, C=1 -> 16
  {
    int total = B * 18 * 96 * 96;
    conv3x3_off_kernel<1, 96, 96>
        <<<(total + 255) / 256, 256, 0, stream>>>(x, w_off1, b_off1, off1, B);
  }
  deform_wmma_kernel<1, 16, 96, 96>
      <<<B * 96 * 96 / 16, 32, 0, stream>>>(x, off1, w1, b1, h1);
  {
    int total = B * 16 * 48 * 48;
    maxpool_kernel<<<(total + 255) / 256, 256, 0, stream>>>(h1, p1, B * 16, 96,
                                                            96);
  }

  // Stage 2: 48x48, C=16 -> 32
  {
    int total = B * 18 * 48 * 48;
    conv3x3_off_kernel<16, 48, 48>
        <<<(total + 255) / 256, 256, 0, stream>>>(p1, w_off2, b_off2, off2, B);
  }
  deform_wmma_kernel<16, 32, 48, 48>
      <<<B * 48 * 48 / 16, 32, 0, stream>>>(p1, off2, w2, b2, h2);
  {
    int total = B * 32 * 24 * 24;
    maxpool_kernel<<<(total + 255) / 256, 256, 0, stream>>>(h2, p2, B * 32, 48,
                                                            48);
  }

  // Stage 3: 24x24, C=32 -> 64
  {
    int total = B * 18 * 24 * 24;
    conv3x3_off_kernel<32, 24, 24>
        <<<(total + 255) / 256, 256, 0, stream>>>(p2, w_off3, b_off3, off3, B);
  }
  deform_wmma_kernel<32, 64, 24, 24>
      <<<B * 24 * 24 / 16, 32, 0, stream>>>(p2, off3, w3, b3, h3);

  // Head
  avgpool_kernel<<<(64 * 64 * 9 + 255) / 256, 256, 0, stream>>>(h3, pooled);
  fc_kernel<<<(640 + 63) / 64, 64, 0, stream>>>(pooled, w_fc, b_fc, out);
}